// DEQLatentSpaceOpt_74217034875067
// MI455X (gfx1250) — compile-verified
//
#include <hip/hip_runtime.h>
#include <hip/hip_bf16.h>

// Problem constants (from reference)
#define T_STEPS 1000
#define NCH     3
#define HH      64
#define WW      64
#define HW      (HH * WW)        // 4096
#define PLANE   (NCH * HW)       // 12288 spatial positions per t
#define NT16    ((T_STEPS + 15) / 16)   // 63 t-tiles of 16

typedef float v2f __attribute__((ext_vector_type(2)));
typedef float v8f __attribute__((ext_vector_type(8)));

// ---------------------------------------------------------------------------
// Kernel 1: E[t, c, h, w] = et_coeff[t] * ( conv3x3(xt)[t,c,h,w] + temb[tidx[t], c] )
// One thread per (t, h, w); computes all 3 output channels so the 27-value
// input patch is loaded once. Zero padding (SAME). Memory-bound; all reads
// are L2-resident (49 MB << 192 MB L2).
// ---------------------------------------------------------------------------
__global__ void conv_et_kernel(const float* __restrict__ xt,
                               const int*   __restrict__ tidx,
                               const float* __restrict__ cw,     // [3][3][3][3]
                               const float* __restrict__ temb,   // [T][3]
                               const float* __restrict__ etc,    // [T]
                               float*       __restrict__ E)      // [T][3][64][64]
{
    int gid = blockIdx.x * blockDim.x + threadIdx.x;
    if (gid >= T_STEPS * HW) return;
    int w = gid & (WW - 1);
    int h = (gid >> 6) & (HH - 1);
    int t = gid >> 12;                       // HW == 4096 == 2^12

    const float* base = xt + (size_t)t * PLANE;
    float acc0 = 0.f, acc1 = 0.f, acc2 = 0.f;
#pragma unroll
    for (int ci = 0; ci < NCH; ++ci) {
#pragma unroll
        for (int kh = 0; kh < 3; ++kh) {
            int hh = h + kh - 1;
            bool hok = (unsigned)hh < (unsigned)HH;
#pragma unroll
            for (int kw = 0; kw < 3; ++kw) {
                int ww2 = w + kw - 1;
                bool ok = hok && ((unsigned)ww2 < (unsigned)WW);
                float v = ok ? base[ci * HW + hh * WW + ww2] : 0.f;
                acc0 = fmaf(cw[((0 * NCH + ci) * 3 + kh) * 3 + kw], v, acc0);
                acc1 = fmaf(cw[((1 * NCH + ci) * 3 + kh) * 3 + kw], v, acc1);
                acc2 = fmaf(cw[((2 * NCH + ci) * 3 + kh) * 3 + kw], v, acc2);
            }
        }
    }
    int   tt = tidx[t];
    float ec = etc[t];
    int   o  = t * PLANE + h * WW + w;
    E[o + 0 * HW] = ec * (acc0 + temb[tt * NCH + 0]);
    E[o + 1 * HW] = ec * (acc1 + temb[tt * NCH + 1]);
    E[o + 2 * HW] = ec * (acc2 + temb[tt * NCH + 2]);
}

// ---------------------------------------------------------------------------
// Kernel 2: blocked cumulative sum over t via fp32 WMMA.
//   P(16t x 16pos) = L(16x16, lower-triangular ones) x E_tile + carry
// realized as 4 chained V_WMMA_F32_16X16X4_F32 (K = 4 per step).
// One wave owns 16 spatial positions and walks all 63 t-tiles serially,
// fusing out[t+1] = alpha_ratio[t]*xT + ps_coeff[t]*prefix.
//
// fp32 WMMA operand layouts (ISA 7.12.2):
//   A 16x4 : lane l -> M = l%16 ; VGPR j holds K = 2*(l/16) + j
//   B 4x16 : lane l -> N = l%16 ; VGPR j holds K = 2*(l/16) + j (mirrored)
//   C/D    : VGPR r, lane l -> element [M = r + 8*(l/16), N = l%16]
// ---------------------------------------------------------------------------
__global__ void scan_wmma_kernel(const float* __restrict__ E,    // [T][12288]
                                 const float* __restrict__ x0,   // xT plane [12288]
                                 const float* __restrict__ ar,   // alpha_ratio [T]
                                 const float* __restrict__ pc,   // et_prevsum_coeff [T]
                                 float*       __restrict__ out)  // [T+1][12288]
{
    int lane  = threadIdx.x & 31;
    int wave  = threadIdx.x >> 5;
    int group = blockIdx.x * (blockDim.x >> 5) + wave;   // 0 .. 767
    int col   = lane & 15;                               // N index (position)
    int half  = lane >> 4;                               // 0 or 1
    int pos   = group * 16 + col;

    float xT = x0[pos];
    if (half == 0) out[pos] = xT;                        // out[0] = xT

    float carry = 0.f;
    for (int b = 0; b < NT16; ++b) {
        int base = b * 16;

        v8f acc;
#pragma unroll
        for (int r = 0; r < 8; ++r) acc[r] = carry;      // C[m,n] = carry[n]

#pragma unroll
        for (int q = 0; q < 4; ++q) {                    // K chunks of 4
            v2f A, B;
#pragma unroll
            for (int j = 0; j < 2; ++j) {
                int kk = 2 * half + j;                   // K within chunk
                int k  = 4 * q + kk;                     // K within 16-tile
                A[j]   = (k <= col) ? 1.0f : 0.0f;       // L[M=col, k]
                int t  = base + k;
                B[j]   = (t < T_STEPS) ? E[t * PLANE + pos] : 0.0f;
            }
            acc = __builtin_amdgcn_wmma_f32_16x16x4_f32(
                      false, A, false, B, (short)0, acc, false, false);
        }

        // carry for next tile = P[15, col] = D vgpr7 on lanes 16..31
        carry = __shfl(acc[7], 16 + col, 32);

#pragma unroll
        for (int r = 0; r < 8; ++r) {
            int t = base + r + 8 * half;                 // M = r + 8*half
            if (t < T_STEPS) {
                out[(t + 1) * PLANE + pos] = ar[t] * xT + pc[t] * acc[r];
            }
        }
    }
}

// ---------------------------------------------------------------------------
// setup_inputs order: 0:x  1:t  2:alpha_ratio  3:et_coeff  4:et_prevsum_coeff
//                     5:conv_w  6:temb
// ---------------------------------------------------------------------------
extern "C" void kernel_launch(void* const* d_in, const int* in_sizes, int n_in,
                              void* d_out, int out_size, void* d_ws, size_t ws_size,
                              hipStream_t stream)
{
    const float* x    = (const float*)d_in[0];
    const int*   tix  = (const int*)  d_in[1];
    const float* ar   = (const float*)d_in[2];
    const float* etc  = (const float*)d_in[3];
    const float* pc   = (const float*)d_in[4];
    const float* cw   = (const float*)d_in[5];
    const float* temb = (const float*)d_in[6];
    float* out = (float*)d_out;
    float* E   = (float*)d_ws;   // needs T*PLANE*4 = 49,152,000 bytes

    const int conv_threads = T_STEPS * HW;               // 4,096,000
    dim3 cblk(256), cgrd((conv_threads + 255) / 256);    // 16000 blocks
    dim3 sblk(256), sgrd((PLANE / 16) / 8);              // 768 waves -> 96 blocks

    for (int it = 0; it < 3; ++it) {
        const float* src = (it == 0) ? x : out;          // xt[:T] from prev result
        conv_et_kernel<<<cgrd, cblk, 0, stream>>>(src, tix, cw, temb, etc, E);
        scan_wmma_kernel<<<sgrd, sblk, 0, stream>>>(E, x, ar, pc, out);
    }
}